// WaveletLoss_60327110640178
// MI455X (gfx1250) — compile-verified
//
#include <hip/hip_runtime.h>

typedef __attribute__((ext_vector_type(4))) float v4f;
typedef __attribute__((ext_vector_type(2))) float v2f;
typedef __attribute__((ext_vector_type(8))) float v8f;

#define THREADS 256
#define WGS_PER_B 256
#define B_COUNT 16
// per-batch geometry: C=3, H=1024, W=1024
// "unit" = one row-pair x 4-column group: 4 B128 loads -> two 2x2 blocks
// units per batch = 3*512*256 = 393216 = ITERS * (WGS_PER_B*THREADS)
#define ITERS 6
#define ROWPAIR_FLOATS 2048u            // 2 rows * 1024 floats
#define ITER_STRIDE_FLOATS (256u * ROWPAIR_FLOATS) // 524288 floats = 2 MB

__device__ __forceinline__ float haar_abs_pair(const v4f& d0, const v4f& d1) {
  // two 2x2 blocks: cols (x,y) and (z,w); |ad|+|da|+|dd| for both (x0.5 deferred)
  float s0 = d0.x + d1.x, s1 = d0.y + d1.y;
  float m0 = d0.x - d1.x, m1 = d0.y - d1.y;
  float r = __builtin_fabsf(s0 - s1) + __builtin_fabsf(m0 + m1) +
            __builtin_fabsf(m0 - m1);
  float s2 = d0.z + d1.z, s3 = d0.w + d1.w;
  float m2 = d0.z - d1.z, m3 = d0.w - d1.w;
  r += __builtin_fabsf(s2 - s3) + __builtin_fabsf(m2 + m3) +
       __builtin_fabsf(m2 - m3);
  return r;
}

__global__ __launch_bounds__(THREADS) void wavelet_main(
    const float* __restrict__ pred, const float* __restrict__ targ,
    float* __restrict__ partial) {
  const unsigned b = blockIdx.y;
  const unsigned tid = threadIdx.x;
  const unsigned gthread = blockIdx.x * THREADS + tid; // 0..65535
  const size_t base = (size_t)b * (3u * 1024u * 1024u);

  // loop-invariant decomposition: wq fixed, row-pair advances 256 per iter
  const unsigned wq = gthread & 255u;  // 4-float column group (coalesced)
  const unsigned rp0 = gthread >> 8;   // starting row-pair (0..255)
  size_t off = base + (size_t)rp0 * ROWPAIR_FLOATS + ((size_t)wq << 2);

  float acc = 0.0f;
  // Compile-time trip count: straight-line body, no exec-mask loop overhead,
  // EXEC provably all-ones for the WMMA below; loads can be pipelined.
#pragma unroll
  for (int i = 0; i < ITERS; ++i) {
    // Streamed once, footprint (402 MB) > L2 (192 MB) -> non-temporal B128.
    v4f p0 = __builtin_nontemporal_load((const v4f*)(pred + off));
    v4f p1 = __builtin_nontemporal_load((const v4f*)(pred + off + 1024u));
    v4f t0 = __builtin_nontemporal_load((const v4f*)(targ + off));
    v4f t1 = __builtin_nontemporal_load((const v4f*)(targ + off + 1024u));
    v4f d0 = p0 - t0;
    v4f d1 = p1 - t1;
    acc += haar_abs_pair(d0, d1);
    off += ITER_STRIDE_FLOATS;
  }

  // --- Wave32 reduction via one V_WMMA_F32_16X16X4_F32 ---
  // A (16x4 f32): lane's VGPR0 = acc, VGPR1 = 0. B = all ones (4x16).
  // D[m][n] = row-sum of A => lanes <16 hold row sums 0-7 in d[0..7],
  // lanes >=16 hold row sums 8-15.
  v2f a;  a.x = acc;  a.y = 0.0f;
  v2f bo; bo.x = 1.0f; bo.y = 1.0f;
  v8f c = {};
  v8f d = __builtin_amdgcn_wmma_f32_16x16x4_f32(
      /*neg_a=*/false, a, /*neg_b=*/false, bo,
      /*c_mod=*/(short)0, c, /*reuse_a=*/false, /*reuse_b=*/false);

  float r = ((d[0] + d[1]) + (d[2] + d[3])) + ((d[4] + d[5]) + (d[6] + d[7]));
  r += __shfl_xor(r, 16, 32); // fold the two half-wave row groups
  // r == sum of all 32 lane accumulators, replicated across the wave.

  __shared__ float sw[THREADS / 32];
  const unsigned lane = tid & 31u, wv = tid >> 5;
  if (lane == 0) sw[wv] = r;
  __syncthreads();
  if (tid == 0) {
    float s = 0.0f;
#pragma unroll
    for (int i = 0; i < THREADS / 32; ++i) s += sw[i];
    partial[b * WGS_PER_B + blockIdx.x] = s;
  }
}

__global__ __launch_bounds__(THREADS) void wavelet_final(
    const float* __restrict__ partial, float* __restrict__ out) {
  const unsigned b = blockIdx.x;
  const unsigned tid = threadIdx.x;
  float v = partial[b * WGS_PER_B + tid];
  // fixed-shape (deterministic) tree reduction
#pragma unroll
  for (int o = 16; o >= 1; o >>= 1) v += __shfl_xor(v, o, 32);
  __shared__ float sw[THREADS / 32];
  if ((tid & 31u) == 0) sw[tid >> 5] = v;
  __syncthreads();
  if (tid == 0) {
    float s = 0.0f;
#pragma unroll
    for (int i = 0; i < THREADS / 32; ++i) s += sw[i];
    // two Haar 1/sqrt(2) factors -> 0.5; mean over 512*512 = 262144 blocks
    out[b] = s * (0.5f / 262144.0f);
  }
}

extern "C" void kernel_launch(void* const* d_in, const int* in_sizes, int n_in,
                              void* d_out, int out_size, void* d_ws,
                              size_t ws_size, hipStream_t stream) {
  const float* pred = (const float*)d_in[0];
  const float* targ = (const float*)d_in[1];
  float* out = (float*)d_out;
  float* partial = (float*)d_ws; // 16 * 256 floats = 16 KB scratch

  dim3 grid(WGS_PER_B, B_COUNT);
  wavelet_main<<<grid, THREADS, 0, stream>>>(pred, targ, partial);
  wavelet_final<<<B_COUNT, THREADS, 0, stream>>>(partial, out);
}